// LinearSumAssignment_34797825032777
// MI455X (gfx1250) — compile-verified
//
#include <hip/hip_runtime.h>
#include <stdint.h>

// ---------------------------------------------------------------------------
// Greedy linear-sum-assignment, 1024 x 16384 fp32, maximize=True.
// Kernel 1: per-row (max value, min col) via TDM tensor_load_to_lds -> LDS.
// Kernel 2: single persistent workgroup runs the 1024 sequential greedy steps
//           on the cached per-row bests; rescans a row only when its best
//           column is consumed (expected ~O(1) rescans per step).
// ---------------------------------------------------------------------------

typedef unsigned int v4u __attribute__((ext_vector_type(4)));
typedef int          v8i __attribute__((ext_vector_type(8)));
typedef int          v4i __attribute__((ext_vector_type(4)));

#define NROWS   1024
#define NCOLS   16384
#define CHUNK   4096            // floats per TDM chunk (16 KB)
#define NEG_INF (-__builtin_inff())

// Issue one TDM DMA: CHUNK contiguous floats from gptr -> LDS byte offset.
// D# group0: count=1 | lds_addr | global_addr[56:0] | type=2.
// D# group1: data_size=4B, tensor_dim0=tile_dim0=CHUNK, tensor_dim1=tile_dim1=1,
//            tensor_dim0_stride=CHUNK (fields are not 32-bit aligned; packed below).
__device__ __forceinline__ void tdm_load_chunk(const float* gptr, unsigned ldsByteOff) {
    uint64_t ga = (uint64_t)(uintptr_t)gptr;
    v4u g0;
    g0.x = 1u;                                                  // count=1, user mode
    g0.y = ldsByteOff;                                          // lds_addr (bytes)
    g0.z = (unsigned)ga;                                        // global_addr[31:0]
    g0.w = (unsigned)((ga >> 32) & 0x01FFFFFFu) | 0x80000000u;  // addr[56:32] | type=2
    v8i g1;
    g1[0] = 0x20000;                 // [17:16] data_size = 2 (4 bytes); wg_mask=0
    g1[1] = (int)(CHUNK << 16);      // bits[63:48] = tensor_dim0[15:0]
    g1[2] = 1 << 16;                 // bits[95:80] = tensor_dim1[15:0] = 1
    g1[3] = (int)(CHUNK << 16);      // bits[127:112] = tile_dim0
    g1[4] = 1;                       // bits[143:128] = tile_dim1 = 1
    g1[5] = CHUNK;                   // bits[191:160] = tensor_dim0_stride[31:0]
    g1[6] = 0;
    g1[7] = 0;
    v4i z4 = {0, 0, 0, 0};
    v8i z8 = {0, 0, 0, 0, 0, 0, 0, 0};
    // clang-23 / therock-10.0 form: (g0, g1, g2, g3, g_extra, cpol)
    __builtin_amdgcn_tensor_load_to_lds(g0, g1, z4, z4, z8, 0);
}

__device__ __forceinline__ void combine(float& mv, int& mc, float v, int c) {
    if (v > mv || (v == mv && c < mc)) { mv = v; mc = c; }
}

// ---------------------------------------------------------------------------
// Kernel 1: one block per row; double-buffered TDM pipeline (2 x 16 KB LDS).
// ---------------------------------------------------------------------------
__global__ __launch_bounds__(256)
void lsa_row_best_kernel(const float* __restrict__ cost,
                         float* __restrict__ bestv,
                         int*   __restrict__ bestc) {
    __shared__ float buf[2][CHUNK];   // static LDS starts at offset 0
    __shared__ float wv[8];
    __shared__ int   wc[8];

    const int tid  = threadIdx.x;
    const int wid  = tid >> 5;
    const int lane = tid & 31;
    const int row  = blockIdx.x;
    const float* rowp = cost + (size_t)row * NCOLS;

    float mv = NEG_INF;
    int   mc = NCOLS;

    // Prime pipeline: chunk0 -> buf0, chunk1 -> buf1. Only wave 0 issues TDM.
    if (wid == 0) {
        tdm_load_chunk(rowp + 0 * CHUNK, 0u);
        tdm_load_chunk(rowp + 1 * CHUNK, (unsigned)(CHUNK * sizeof(float)));
        __builtin_amdgcn_s_wait_tensorcnt(1);    // chunk0 landed
    }
    __syncthreads();
    // reduce chunk0 (cols ascending per thread -> '>' keeps min col; use full key anyway)
    #pragma unroll
    for (int j = 0; j < CHUNK / 256; ++j) {
        int idx = tid + j * 256;
        combine(mv, mc, buf[0][idx], 0 * CHUNK + idx);
    }
    __syncthreads();                              // buf0 free for reuse

    if (wid == 0) {
        tdm_load_chunk(rowp + 2 * CHUNK, 0u);     // chunk2 -> buf0
        __builtin_amdgcn_s_wait_tensorcnt(1);     // chunk1 landed
    }
    __syncthreads();
    #pragma unroll
    for (int j = 0; j < CHUNK / 256; ++j) {
        int idx = tid + j * 256;
        combine(mv, mc, buf[1][idx], 1 * CHUNK + idx);
    }
    __syncthreads();                              // buf1 free for reuse

    if (wid == 0) {
        tdm_load_chunk(rowp + 3 * CHUNK, (unsigned)(CHUNK * sizeof(float)));  // chunk3 -> buf1
        __builtin_amdgcn_s_wait_tensorcnt(1);     // chunk2 landed
    }
    __syncthreads();
    #pragma unroll
    for (int j = 0; j < CHUNK / 256; ++j) {
        int idx = tid + j * 256;
        combine(mv, mc, buf[0][idx], 2 * CHUNK + idx);
    }
    __syncthreads();

    if (wid == 0) __builtin_amdgcn_s_wait_tensorcnt(0);  // chunk3 landed
    __syncthreads();
    #pragma unroll
    for (int j = 0; j < CHUNK / 256; ++j) {
        int idx = tid + j * 256;
        combine(mv, mc, buf[1][idx], 3 * CHUNK + idx);
    }

    // wave32 shuffle reduce, then 8 wave partials -> thread 0
    #pragma unroll
    for (int m = 16; m; m >>= 1) {
        float ov = __shfl_xor(mv, m, 32);
        int   oc = __shfl_xor(mc, m, 32);
        combine(mv, mc, ov, oc);
    }
    if (lane == 0) { wv[wid] = mv; wc[wid] = mc; }
    __syncthreads();
    if (tid == 0) {
        float bmv = wv[0]; int bmc = wc[0];
        #pragma unroll
        for (int i = 1; i < 8; ++i) combine(bmv, bmc, wv[i], wc[i]);
        bestv[row] = bmv;
        bestc[row] = bmc;
    }
}

// ---------------------------------------------------------------------------
// Kernel 2: 1 block x 1024 threads (32 waves) does the 1024 greedy steps.
// ---------------------------------------------------------------------------
__global__ __launch_bounds__(1024)
void lsa_greedy_kernel(const float* __restrict__ cost,
                       const float* __restrict__ bestv_g,
                       const int*   __restrict__ bestc_g,
                       float* __restrict__ out) {
    __shared__ float    bv[NROWS];
    __shared__ int      bc[NROWS];
    __shared__ unsigned colUsed[NCOLS / 32];
    __shared__ float    wredv[32];
    __shared__ int      wredi[32];
    __shared__ int      assignedCol[NROWS];
    __shared__ int      invalList[NROWS];
    __shared__ int      invalCount;
    __shared__ int      curC;

    const int tid  = threadIdx.x;
    const int wid  = tid >> 5;
    const int lane = tid & 31;

    bv[tid] = bestv_g[tid];
    bc[tid] = bestc_g[tid];
    if (tid < NCOLS / 32) colUsed[tid] = 0u;
    __syncthreads();

    for (int step = 0; step < NROWS; ++step) {
        // --- phase 1: argmax over per-row bests (tie -> min row == min flat idx)
        float v = bv[tid];
        int   i = tid;
        #pragma unroll
        for (int m = 16; m; m >>= 1) {
            float ov = __shfl_xor(v, m, 32);
            int   oi = __shfl_xor(i, m, 32);
            if (ov > v || (ov == v && oi < i)) { v = ov; i = oi; }
        }
        if (lane == 0) { wredv[wid] = v; wredi[wid] = i; }
        __syncthreads();
        if (wid == 0) {
            float v2 = wredv[lane];
            int   i2 = wredi[lane];
            #pragma unroll
            for (int m = 16; m; m >>= 1) {
                float ov = __shfl_xor(v2, m, 32);
                int   oi = __shfl_xor(i2, m, 32);
                if (ov > v2 || (ov == v2 && oi < i2)) { v2 = ov; i2 = oi; }
            }
            if (lane == 0) {
                int r = i2;
                int c = bc[r];
                curC = c;
                assignedCol[r] = c;
                colUsed[c >> 5] |= (1u << (c & 31));
                bv[r] = NEG_INF;          // row consumed
                invalCount = 0;
            }
        }
        __syncthreads();

        // --- phase 2: collect live rows whose cached best col was just taken
        const int c = curC;
        if (bv[tid] != NEG_INF && bc[tid] == c) {
            int p = atomicAdd(&invalCount, 1);
            invalList[p] = tid;
        }
        __syncthreads();

        // --- phase 3: one wave per stale row rescans it over unused cols
        const int n = invalCount;
        for (int j = wid; j < n; j += 32) {
            const int row = invalList[j];
            const float* rp = cost + (size_t)row * NCOLS;
            float mv = NEG_INF;
            int   mc = NCOLS;
            for (int col = lane; col < NCOLS; col += 32) {
                if (!((colUsed[col >> 5] >> (col & 31)) & 1u)) {
                    float cv = rp[col];
                    if (cv > mv) { mv = cv; mc = col; }   // ascending col -> min-col tie kept
                }
            }
            #pragma unroll
            for (int m = 16; m; m >>= 1) {
                float ov = __shfl_xor(mv, m, 32);
                int   oc = __shfl_xor(mc, m, 32);
                if (ov > mv || (ov == mv && oc < mc)) { mv = ov; mc = oc; }
            }
            if (lane == 0) { bv[row] = mv; bc[row] = mc; }
        }
        __syncthreads();
    }

    // Output: (row_ind sorted = identity, col_ind per row), concatenated flat.
    out[tid]         = (float)tid;
    out[NROWS + tid] = (float)assignedCol[tid];
}

// ---------------------------------------------------------------------------
extern "C" void kernel_launch(void* const* d_in, const int* in_sizes, int n_in,
                              void* d_out, int out_size, void* d_ws, size_t ws_size,
                              hipStream_t stream) {
    const float* cost = (const float*)d_in[0];
    float* bestv = (float*)d_ws;
    int*   bestc = (int*)((char*)d_ws + NROWS * sizeof(float));

    lsa_row_best_kernel<<<NROWS, 256, 0, stream>>>(cost, bestv, bestc);
    lsa_greedy_kernel<<<1, NROWS, 0, stream>>>(cost, bestv, bestc, (float*)d_out);
}